// InstantNGP_41867341201408
// MI455X (gfx1250) — compile-verified
//
#include <hip/hip_runtime.h>
#include <cstdint>
#include <cstddef>

// ---------------------------------------------------------------------------
// Instant-NGP encode for MI455X (gfx1250, wave32)
//   out = [ enc_pts (4096*128*32) | enc_dirs (4096*128*16) | dists (4096*128) ]
// ---------------------------------------------------------------------------

#define NRAYS     4096
#define NSAMP     128
#define NLEVELS   16
#define TBL_SIZE  524288u        // 2^19 entries per level
#define TBL_MASK  0x7FFFFu
#define PRIME1    2654435761u
#define PRIME2    805459861u

#define ENC_PTS_ELEMS  ((size_t)NRAYS * NSAMP * 32)   // 16777216
#define ENC_DIRS_OFF   ENC_PTS_ELEMS
#define DIST_OFF       (ENC_PTS_ELEMS + (size_t)NRAYS * NSAMP * 16)  // 25165824

__constant__ int RES_TBL[NLEVELS] = {16, 22, 30, 42, 58, 80, 111, 153,
                                     212, 294, 406, 561, 776, 1072, 1482, 2048};

// ------------------------- Kernel 1: hash grid + dists ----------------------

__device__ __forceinline__ void level_feat(const float2* __restrict__ tab,
                                           float res, float px, float py, float pz,
                                           float& f0, float& f1) {
    float cell = 16.0f / res;
    float gx = floorf((px + 8.0f) / cell);
    float gy = floorf((py + 8.0f) / cell);
    float gz = floorf((pz + 8.0f) / cell);
    float dx = (px - (gx * cell - 8.0f)) / cell;
    float dy = (py - (gy * cell - 8.0f)) / cell;
    float dz = (pz - (gz * cell - 8.0f)) / cell;
    uint32_t ix = (uint32_t)(int)gx, iy = (uint32_t)(int)gy, iz = (uint32_t)(int)gz;
    uint32_t hx0 = ix,              hx1 = ix + 1u;
    uint32_t hy0 = iy * PRIME1,     hy1 = (iy + 1u) * PRIME1;
    uint32_t hz0 = iz * PRIME2,     hz1 = (iz + 1u) * PRIME2;
    // 8 corners in VERTS order (x,y,z): 000,001,010,011,100,101,110,111
    float2 e0 = tab[(hx0 ^ hy0 ^ hz0) & TBL_MASK];
    float2 e1 = tab[(hx0 ^ hy0 ^ hz1) & TBL_MASK];
    float2 e2 = tab[(hx0 ^ hy1 ^ hz0) & TBL_MASK];
    float2 e3 = tab[(hx0 ^ hy1 ^ hz1) & TBL_MASK];
    float2 e4 = tab[(hx1 ^ hy0 ^ hz0) & TBL_MASK];
    float2 e5 = tab[(hx1 ^ hy0 ^ hz1) & TBL_MASK];
    float2 e6 = tab[(hx1 ^ hy1 ^ hz0) & TBL_MASK];
    float2 e7 = tab[(hx1 ^ hy1 ^ hz1) & TBL_MASK];
    float wx0 = 1.0f - dx, wy0 = 1.0f - dy, wz0 = 1.0f - dz;
    float w0 = wx0 * wy0 * wz0, w1 = wx0 * wy0 * dz;
    float w2 = wx0 * dy * wz0,  w3 = wx0 * dy * dz;
    float w4 = dx * wy0 * wz0,  w5 = dx * wy0 * dz;
    float w6 = dx * dy * wz0,   w7 = dx * dy * dz;
    float a = w0 * e0.x, b = w0 * e0.y;
    a = fmaf(w1, e1.x, a); b = fmaf(w1, e1.y, b);
    a = fmaf(w2, e2.x, a); b = fmaf(w2, e2.y, b);
    a = fmaf(w3, e3.x, a); b = fmaf(w3, e3.y, b);
    a = fmaf(w4, e4.x, a); b = fmaf(w4, e4.y, b);
    a = fmaf(w5, e5.x, a); b = fmaf(w5, e5.y, b);
    a = fmaf(w6, e6.x, a); b = fmaf(w6, e6.y, b);
    a = fmaf(w7, e7.x, a); b = fmaf(w7, e7.y, b);
    f0 = a; f1 = b;
}

__global__ __launch_bounds__(NSAMP)
void ngp_hash_dists(const float* __restrict__ rays,
                    const float* __restrict__ t_rand,
                    const float* __restrict__ tables,
                    float* __restrict__ out) {
    const int r = blockIdx.x;      // ray (uniform per block -> scalar loads)
    const int s = threadIdx.x;     // sample

    const float dxr = rays[r * 6 + 0], dyr = rays[r * 6 + 1], dzr = rays[r * 6 + 2];
    const float oxr = rays[r * 6 + 3], oyr = rays[r * 6 + 4], ozr = rays[r * 6 + 5];

    const float step = 4.0f / 127.0f;                 // linspace(2,6,128) step
    const float ts   = fmaf((float)s, step, 2.0f);    // t[s]
    const float px = fmaf(dxr, ts, oxr);
    const float py = fmaf(dyr, ts, oyr);
    const float pz = fmaf(dzr, ts, ozr);

    // ---- hash-grid features: 2 levels per iteration, one b128 store each --
    const size_t base = ((size_t)r * NSAMP + (size_t)s) * 32;
    float4* __restrict__ dst4 = (float4*)(out + base);
    for (int ii = 0; ii < 8; ++ii) {
        const int l0 = 2 * ii, l1 = 2 * ii + 1;
        float a0, a1, b0, b1;
        level_feat((const float2*)tables + (size_t)l0 * TBL_SIZE,
                   (float)RES_TBL[l0], px, py, pz, a0, a1);
        level_feat((const float2*)tables + (size_t)l1 * TBL_SIZE,
                   (float)RES_TBL[l1], px, py, pz, b0, b1);
        dst4[ii] = make_float4(a0, a1, b0, b1);
    }

    // ---- dists (jittered t deltas) ----------------------------------------
    const float tsm1 = fmaf((float)(s - 1), step, 2.0f);
    const float tsp1 = fmaf((float)(s + 1), step, 2.0f);
    const float tsp2 = fmaf((float)(s + 2), step, 2.0f);
    const float lower = (s == 0)   ? ts : 0.5f * (tsm1 + ts);
    const float upper = (s == 127) ? ts : 0.5f * (ts + tsp1);
    const float tj = fmaf(upper - lower, t_rand[(size_t)r * NSAMP + s], lower);
    float dist;
    if (s == 127) {
        dist = 1e10f;
    } else {
        const float lo1 = 0.5f * (ts + tsp1);
        const float up1 = (s + 1 == 127) ? tsp1 : 0.5f * (tsp1 + tsp2);
        const float tj1 = fmaf(up1 - lo1, t_rand[(size_t)r * NSAMP + s + 1], lo1);
        dist = tj1 - tj;
    }
    out[DIST_OFF + (size_t)r * NSAMP + s] = dist;
}

// -------------- Kernel 2: SH deg-3 via V_WMMA_F32_16X16X4_F32 ---------------
// SH[ray][comp] = sum_k mono_k(dir_ray) * BCO[k][comp], 20 monomials, 16 comps.
// 16 rays per wave tile; 5 chained WMMAs (K=4 each).

typedef __attribute__((ext_vector_type(2))) float v2f;
typedef __attribute__((ext_vector_type(8))) float v8f;

#define SH_C0  0.28209479177387814
#define SH_C1  0.4886025119029199
#define C20    1.0925484305920792
#define C21   (-1.0925484305920792)
#define C22    0.31539156525252005
#define C23   (-1.0925484305920792)
#define C24    0.5462742152960396
#define C30   (-0.5900435899266435)
#define C31    2.890611442640554
#define C32   (-0.4570457994644658)
#define C33    0.3731763325901154
#define C34   (-0.4570457994644658)
#define C35    1.445305721320277
#define C36   (-0.5900435899266435)

// Monomial order k: 1,x,y,z, x2,y2,z2,xy, yz,xz,x2y,y3, xyz,yz2,x2z,y2z, z3,xz2,x3,xy2
__constant__ float BCO[20][16] = {
 /* 1  */ {(float)SH_C0,0,0,0, 0,0,0,0, 0,0,0,0, 0,0,0,0},
 /* x  */ {0,0,0,(float)(-SH_C1), 0,0,0,0, 0,0,0,0, 0,0,0,0},
 /* y  */ {0,(float)(-SH_C1),0,0, 0,0,0,0, 0,0,0,0, 0,0,0,0},
 /* z  */ {0,0,(float)SH_C1,0, 0,0,0,0, 0,0,0,0, 0,0,0,0},
 /* x2 */ {0,0,0,0, 0,0,(float)(-C22),0, (float)C24,0,0,0, 0,0,0,0},
 /* y2 */ {0,0,0,0, 0,0,(float)(-C22),0, (float)(-C24),0,0,0, 0,0,0,0},
 /* z2 */ {0,0,0,0, 0,0,(float)(2.0*C22),0, 0,0,0,0, 0,0,0,0},
 /* xy */ {0,0,0,0, (float)C20,0,0,0, 0,0,0,0, 0,0,0,0},
 /* yz */ {0,0,0,0, 0,(float)C21,0,0, 0,0,0,0, 0,0,0,0},
 /* xz */ {0,0,0,0, 0,0,0,(float)C23, 0,0,0,0, 0,0,0,0},
 /* x2y*/ {0,0,0,0, 0,0,0,0, 0,(float)(3.0*C30),0,(float)(-C32), 0,0,0,0},
 /* y3 */ {0,0,0,0, 0,0,0,0, 0,(float)(-C30),0,(float)(-C32), 0,0,0,0},
 /* xyz*/ {0,0,0,0, 0,0,0,0, 0,0,(float)C31,0, 0,0,0,0},
 /* yz2*/ {0,0,0,0, 0,0,0,0, 0,0,0,(float)(4.0*C32), 0,0,0,0},
 /* x2z*/ {0,0,0,0, 0,0,0,0, 0,0,0,0, (float)(-3.0*C33),0,(float)C35,0},
 /* y2z*/ {0,0,0,0, 0,0,0,0, 0,0,0,0, (float)(-3.0*C33),0,(float)(-C35),0},
 /* z3 */ {0,0,0,0, 0,0,0,0, 0,0,0,0, (float)(2.0*C33),0,0,0},
 /* xz2*/ {0,0,0,0, 0,0,0,0, 0,0,0,0, 0,(float)(4.0*C34),0,0},
 /* x3 */ {0,0,0,0, 0,0,0,0, 0,0,0,0, 0,(float)(-C34),0,(float)C36},
 /* xy2*/ {0,0,0,0, 0,0,0,0, 0,0,0,0, 0,(float)(-C34),0,(float)(-3.0*C36)},
};

__global__ __launch_bounds__(256)
void ngp_sh_wmma(const float* __restrict__ rays, float* __restrict__ enc_dirs) {
    __shared__ float sh[16 * 16];   // [ray_in_tile][component]
    const int tid = threadIdx.x;
    const int rayBase = blockIdx.x * 16;

    if (tid < 32) {               // wave 0 only (wave-uniform branch, EXEC all 1s)
        const int n  = tid & 15;  // A row (ray in tile) == B column (SH comp)
        const int hi = tid >> 4;  // lane half selects K pair {0,1} vs {2,3}
        const float* rp = rays + (size_t)(rayBase + n) * 6;
        const float x = rp[0], y = rp[1], z = rp[2];
        const float one = 1.0f;
        const float xx = x * x, yy = y * y, zz = z * z;
        const float xy = x * y, yz = y * z, xz = x * z;
        const float xxy = xx * y, yyy = yy * y, xyz = xy * z, yzz = yz * z;
        const float xxz = xx * z, yyz = yy * z, zzz = zz * z, xzz = xz * z;
        const float xxx = xx * x, xyy = x * yy;
        const int khalf = hi * 2;
        v8f acc = {0.f, 0.f, 0.f, 0.f, 0.f, 0.f, 0.f, 0.f};

#define SH_CHUNK(c, m0_, m1_, m2_, m3_) {                                   \
        v2f a, b;                                                           \
        a.x = hi ? (m2_) : (m0_);                                           \
        a.y = hi ? (m3_) : (m1_);                                           \
        const int kb = 4 * (c) + khalf;                                     \
        b.x = BCO[kb][n];                                                   \
        b.y = BCO[kb + 1][n];                                               \
        acc = __builtin_amdgcn_wmma_f32_16x16x4_f32(                        \
            false, a, false, b, (short)0, acc, false, false);               \
    }
        SH_CHUNK(0, one, x,   y,   z)
        SH_CHUNK(1, xx,  yy,  zz,  xy)
        SH_CHUNK(2, yz,  xz,  xxy, yyy)
        SH_CHUNK(3, xyz, yzz, xxz, yyz)
        SH_CHUNK(4, zzz, xzz, xxx, xyy)
#undef SH_CHUNK

        // D layout: lane holds D[M][N], M = v + 8*hi, N = lane&15
        const int row0 = hi * 8;
#pragma unroll
        for (int v = 0; v < 8; ++v) sh[(row0 + v) * 16 + n] = acc[v];
    }
    __syncthreads();

    // Broadcast each ray's 16 SH comps to its 128 samples; coalesced b128.
    for (int it = tid; it < 16 * 128 * 4; it += 256) {
        const int ray_in = it >> 9;        // 512 float4 per ray
        const int rem    = it & 511;
        const int sample = rem >> 2;
        const int q      = rem & 3;
        const float4 val = *(const float4*)&sh[ray_in * 16 + q * 4];
        const size_t dst = (((size_t)(rayBase + ray_in)) * NSAMP + (size_t)sample) * 16
                           + (size_t)(q * 4);
        *(float4*)(enc_dirs + dst) = val;
    }
}

// --------------------------------- launch -----------------------------------

extern "C" void kernel_launch(void* const* d_in, const int* in_sizes, int n_in,
                              void* d_out, int out_size, void* d_ws, size_t ws_size,
                              hipStream_t stream) {
    const float* rays   = (const float*)d_in[0];   // (4096, 6)
    const float* t_rand = (const float*)d_in[1];   // (4096, 128)
    const float* tables = (const float*)d_in[2];   // (16, 2^19, 2)
    float* out = (float*)d_out;

    ngp_hash_dists<<<NRAYS, NSAMP, 0, stream>>>(rays, t_rand, tables, out);
    ngp_sh_wmma<<<NRAYS / 16, 256, 0, stream>>>(rays, out + ENC_DIRS_OFF);
}